// Npool_67190468378665
// MI455X (gfx1250) — compile-verified
//
#include <hip/hip_runtime.h>

// Shapes from the reference: x is (8, 32, 64, 4096) f32; pool=4, stride=2, nn=1.
#define ROWW    4096
#define OUTW    ((ROWW - 4) / 2 + 1)             // 2047
#define TPB     256                              // 8 wave32s per workgroup

// Branchless window op: first-occurrence argmax over {w0..w3}, then
//   mi in {0,3} (edge)   -> max
//   mi in {1,2} (inner)  -> 0.25*left + 0.5*max + 0.25*right
__device__ __forceinline__ float npool_win(float w0, float w1, float w2, float w3) {
    bool  b1 = w1 > w0;
    float m  = b1 ? w1 : w0;
    int   mi = b1 ? 1 : 0;
    bool  b2 = w2 > m;  m = b2 ? w2 : m;  mi = b2 ? 2 : mi;
    bool  b3 = w3 > m;  m = b3 ? w3 : m;  mi = b3 ? 3 : mi;
    const float l     = (mi == 1) ? w0 : w1;
    const float r     = (mi == 1) ? w2 : w3;
    const float inner = 0.5f * m + 0.25f * (l + r);
    const bool  edge  = (mi == 0) || (mi == 3);
    return edge ? m : inner;
}

__global__ __launch_bounds__(TPB)
void npool_kernel(const float* __restrict__ x, float* __restrict__ out) {
    __shared__ __align__(16) float smem[ROWW];   // 16 KB row staging buffer

    const unsigned tid  = threadIdx.x;
    const float*   rowp = x + (size_t)blockIdx.x * ROWW;

    // ---- Stage the full input row into LDS via CDNA5 async copies (ASYNCcnt).
    // saddr form: global addr = SGPR64(rowp) + VGPR32(byte off). LDS destination
    // is the low 32 bits of the generic pointer (= wave-relative LDS byte addr);
    // the ptrtoint capture keeps smem alive under the "memory" clobber.
#pragma unroll
    for (int k = 0; k < ROWW / (4 * TPB); ++k) {         // 4 iterations
        const unsigned foff  = (unsigned)(k * TPB + tid) * 4u;  // float index of this b128
        const unsigned gboff = foff * 4u;                       // global byte offset
        const unsigned long long lp64 = (unsigned long long)&smem[foff];
        const unsigned ldsaddr = (unsigned)lp64;                // LDS byte address (VDST)
        asm volatile("global_load_async_to_lds_b128 %0, %1, %2"
                     :: "v"(ldsaddr), "v"(gboff), "s"(rowp)
                     : "memory");
    }
    // Drain this wave's async copies, then barrier so every wave sees the whole row.
    asm volatile("s_wait_asynccnt 0" ::: "memory");
    __syncthreads();

    // ---- Compute: thread handles output pair (2t, 2t+1) from smem[4t .. 4t+5]:
    // one 16B-aligned ds_load_b128 + one ds_load_b64 per pair (6 floats / 2 outputs).
    float* orow = out + (size_t)blockIdx.x * OUTW;
#pragma unroll
    for (int k = 0; k < 4; ++k) {
        const int t = k * TPB + (int)tid;                // 0 .. 1023
        if (t < (OUTW / 2)) {                            // 1023 full pairs
            const float4 q = *reinterpret_cast<const float4*>(&smem[4 * t]);
            const float2 e = *reinterpret_cast<const float2*>(&smem[4 * t + 4]);
            const float r0 = npool_win(q.x, q.y, q.z, q.w);
            const float r1 = npool_win(q.z, q.w, e.x, e.y);
            __builtin_nontemporal_store(r0, &orow[2 * t]);
            __builtin_nontemporal_store(r1, &orow[2 * t + 1]);
        } else if (t == (OUTW / 2)) {                    // tail output o = 2046
            const float4 q = *reinterpret_cast<const float4*>(&smem[ROWW - 4]);
            __builtin_nontemporal_store(npool_win(q.x, q.y, q.z, q.w),
                                        &orow[OUTW - 1]);
        }
    }
}

extern "C" void kernel_launch(void* const* d_in, const int* in_sizes, int n_in,
                              void* d_out, int out_size, void* d_ws, size_t ws_size,
                              hipStream_t stream) {
    const float* x   = (const float*)d_in[0];
    float*       out = (float*)d_out;
    const int rows   = in_sizes[0] / ROWW;   // 8*32*64 = 16384
    npool_kernel<<<dim3(rows), dim3(TPB), 0, stream>>>(x, out);
}